// RelationalDelayGNNStage_28784870818304
// MI455X (gfx1250) — compile-verified
//
#include <hip/hip_runtime.h>
#include <stdint.h>

// ---------------------------------------------------------------------------
// RelationalDelayGNNStage for MI455X (gfx1250, wave32, WMMA + TDM)
//
// Per timestep t (t = 0..3):
//   acc = 0
//   for e in 0..2:   h = xs[t]       @ W_edge[t,e] + b_edge[t,e]; scatter(etype==e)
//   for k in 2..t+1: h = xs[t-(k-1)] @ W_hop[t,k-2] + b_hop[t,k-2]; scatter(hop==k)
//   xs[t+1] = xs[t] + relu(acc)      (t==3 -> d_out)
//
// GEMM: V_WMMA_F32_16X16X4_F32, B-matrix staged into LDS by the Tensor Data
// Mover (tensor_load_to_lds, TENSORcnt), double-buffered 32x256 fp32 k-slabs.
// Edge pass: per-warp gather + global_atomic_add_f32 scatter (L2-resident).
// Workspace: acc | h | xs1 | xs2 | xs3 = 5 * N*D*4 B = 102.4 MB.
// ---------------------------------------------------------------------------

#define DD 256
#define KCHUNK 32
#define NCHUNKS (DD / KCHUNK)

typedef float v2f __attribute__((ext_vector_type(2)));
typedef float v8f __attribute__((ext_vector_type(8)));
typedef unsigned int u32x4 __attribute__((ext_vector_type(4)));
typedef int i32x4 __attribute__((ext_vector_type(4)));
typedef int i32x8 __attribute__((ext_vector_type(8)));

// Issue a TDM load of a (rows x 256) fp32 tile of W (row stride 256 elems)
// into LDS at byte offset ldsOff. D# layout per CDNA5 ISA §8.3/8.4.
__device__ __forceinline__ void tdm_load_w_chunk(const float* gsrc, uint32_t ldsOff, int rows)
{
    uint64_t ga = (uint64_t)(uintptr_t)gsrc;
    u32x4 g0;
    g0[0] = 1u;                                        // count=1, user mode, no gather
    g0[1] = ldsOff;                                    // lds_addr (bytes)
    g0[2] = (uint32_t)(ga & 0xffffffffu);              // global_addr[31:0]  (bits 95:64)
    g0[3] = (uint32_t)((ga >> 32) & 0x1ffffffu)        // global_addr[56:32] (bits 120:96)
          | (2u << 30);                                // type=2 (bits 127:126)
    i32x8 g1;
    g1[0] = 0x00020000;                                // wg_mask=0, data_size=2 (4 B)
    g1[1] = (int)(256u << 16);                         // tensor_dim0[15:0]=256 @ bits 63:48
    g1[2] = (int)(256u << 16);                         // tensor_dim0 hi=0 | tensor_dim1 lo=256
    g1[3] = (int)(256u << 16);                         // tensor_dim1 hi=0 | tile_dim0=256
    g1[4] = rows & 0xffff;                             // tile_dim1=rows, tile_dim2=0
    g1[5] = 256;                                       // tensor_dim0_stride low (elems)
    g1[6] = 0;
    g1[7] = 0;
    i32x4 gz = {0, 0, 0, 0};
#if defined(__clang_major__) && (__clang_major__ >= 23)
    i32x8 gz8 = {0, 0, 0, 0, 0, 0, 0, 0};
    __builtin_amdgcn_tensor_load_to_lds(g0, g1, gz, gz, gz8, 0);
#else
    __builtin_amdgcn_tensor_load_to_lds(g0, g1, gz, gz, 0);
#endif
}

// H[N x 256] = X[N x 256] @ W[256 x 256] + bias[256]
// Block = 256 threads = 8 waves; block tile 32 rows x 256 cols.
// Wave (msub, n0) = ((w>>2)*16, (w&3)*64); each wave: 4 WMMA accumulators.
__global__ __launch_bounds__(256) void gemm_bias_wmma(
    const float* __restrict__ X,
    const float* __restrict__ W,      // row major: W[k*256 + n]
    const float* __restrict__ bias,
    float* __restrict__ H)
{
    __shared__ float Wlds[2 * KCHUNK * DD];            // 2 x 32 KB, double buffer

    const int tid  = threadIdx.x;
    const int lane = tid & 31;
    const int wave = tid >> 5;                         // 0..7
    const int m0   = blockIdx.x << 5;                  // 32 rows / block
    const int msub = (wave >> 2) << 4;                 // 0 or 16
    const int n0   = (wave & 3) << 6;                  // 0,64,128,192
    const int hi   = lane >> 4;                        // half-wave: K pair select
    const int lo   = lane & 15;

    const float* xrow = X + (size_t)(m0 + msub + lo) * DD;
    const uint32_t ldsBase = (uint32_t)(uintptr_t)(&Wlds[0]);

    v8f acc0 = {}, acc1 = {}, acc2 = {}, acc3 = {};

    if (wave == 0) {
        tdm_load_w_chunk(W, ldsBase, KCHUNK);          // prologue: chunk 0
    }

    for (int c = 0; c < NCHUNKS; ++c) {
        if (wave == 0) {
            __builtin_amdgcn_s_wait_tensorcnt(0);      // chunk c landed in LDS
        }
        __syncthreads();                               // chunk c visible; buf (c+1)&1 free
        if (wave == 0 && (c + 1) < NCHUNKS) {
            tdm_load_w_chunk(W + (size_t)(c + 1) * KCHUNK * DD,
                             ldsBase + (uint32_t)(((c + 1) & 1) * (KCHUNK * DD * 4)),
                             KCHUNK);
        }

        const float* Wc = &Wlds[(c & 1) * (KCHUNK * DD)];
        const int kc = c * KCHUNK;

#pragma unroll
        for (int kk = 0; kk < KCHUNK; kk += 4) {
            // A fragment (16x4 fp32): lanes 0-15 K={k,k+1}, lanes 16-31 K={k+2,k+3}
            v2f a;
            a.x = xrow[kc + kk + 2 * hi + 0];
            a.y = xrow[kc + kk + 2 * hi + 1];
            // B fragments (4x16 fp32) from LDS
            const float* wr0 = Wc + (kk + 2 * hi) * DD + n0 + lo;
            const float* wr1 = wr0 + DD;
            v2f b0, b1, b2, b3;
            b0.x = wr0[0];  b0.y = wr1[0];
            b1.x = wr0[16]; b1.y = wr1[16];
            b2.x = wr0[32]; b2.y = wr1[32];
            b3.x = wr0[48]; b3.y = wr1[48];

            acc0 = __builtin_amdgcn_wmma_f32_16x16x4_f32(false, a, false, b0, (short)0, acc0, false, false);
            acc1 = __builtin_amdgcn_wmma_f32_16x16x4_f32(false, a, false, b1, (short)0, acc1, false, false);
            acc2 = __builtin_amdgcn_wmma_f32_16x16x4_f32(false, a, false, b2, (short)0, acc2, false, false);
            acc3 = __builtin_amdgcn_wmma_f32_16x16x4_f32(false, a, false, b3, (short)0, acc3, false, false);
        }
        __syncthreads();                               // all consumed chunk c before reuse
    }

    // C/D layout: lane (hi,lo): VGPR r -> (row = m0+msub + r + 8*hi, col = n_tile + lo)
#pragma unroll
    for (int j = 0; j < 4; ++j) {
        v8f accv = (j == 0) ? acc0 : (j == 1) ? acc1 : (j == 2) ? acc2 : acc3;
        const int col = n0 + j * 16 + lo;
        const float bv = bias[col];
#pragma unroll
        for (int r = 0; r < 8; ++r) {
            const int row = m0 + msub + r + 8 * hi;
            H[(size_t)row * DD + col] = accv[r] + bv;
        }
    }
}

// One warp per edge: if attr[2e+field]==val, acc[dst] += h[src] (256 floats).
__global__ __launch_bounds__(256) void edge_scatter(
    const float* __restrict__ h, float* __restrict__ acc,
    const int* __restrict__ src, const int* __restrict__ dst,
    const int* __restrict__ attr, int E, int field, int val)
{
    const int lane   = threadIdx.x & 31;
    const int warp   = (blockIdx.x * blockDim.x + threadIdx.x) >> 5;
    const int nwarps = (gridDim.x * blockDim.x) >> 5;

    for (int e = warp; e < E; e += nwarps) {
        if (attr[2 * e + field] != val) continue;      // warp-uniform branch
        const int s = src[e];
        const int d = dst[e];
        const float4* hp = (const float4*)(h + (size_t)s * DD);
        float4 v0 = hp[lane * 2 + 0];
        float4 v1 = hp[lane * 2 + 1];
        float* ap = acc + (size_t)d * DD + lane * 8;
        atomicAdd(ap + 0, v0.x); atomicAdd(ap + 1, v0.y);
        atomicAdd(ap + 2, v0.z); atomicAdd(ap + 3, v0.w);
        atomicAdd(ap + 4, v1.x); atomicAdd(ap + 5, v1.y);
        atomicAdd(ap + 6, v1.z); atomicAdd(ap + 7, v1.w);
    }
}

__global__ __launch_bounds__(256) void zero_f4(float4* __restrict__ p, int n4)
{
    int i = blockIdx.x * blockDim.x + threadIdx.x;
    if (i < n4) p[i] = make_float4(0.f, 0.f, 0.f, 0.f);
}

__global__ __launch_bounds__(256) void relu_add(
    const float* __restrict__ x, const float* __restrict__ acc,
    float* __restrict__ out, int n)
{
    int i = blockIdx.x * blockDim.x + threadIdx.x;
    if (i < n) {
        float a = acc[i];
        out[i] = x[i] + (a > 0.f ? a : 0.f);
    }
}

extern "C" void kernel_launch(void* const* d_in, const int* in_sizes, int n_in,
                              void* d_out, int out_size, void* d_ws, size_t ws_size,
                              hipStream_t stream)
{
    (void)n_in; (void)out_size; (void)ws_size;

    const float* x      = (const float*)d_in[0];   // N x 256
    const int*   eidx   = (const int*)  d_in[1];   // 2 x E
    const int*   attr   = (const int*)  d_in[2];   // E x 2  (hop, etype)
    const float* W_edge = (const float*)d_in[3];   // 4 x 3 x 256 x 256
    const float* b_edge = (const float*)d_in[4];   // 4 x 3 x 256
    const float* W_hop  = (const float*)d_in[5];   // 4 x 3 x 256 x 256
    const float* b_hop  = (const float*)d_in[6];   // 4 x 3 x 256
    float* out = (float*)d_out;

    const int ND = in_sizes[0];       // N*D
    const int N  = ND / DD;           // 20000
    const int E  = in_sizes[1] / 2;   // 500000
    const int* src = eidx;
    const int* dst = eidx + E;

    // Workspace layout (floats): acc | h | xs1 | xs2 | xs3
    float* acc = (float*)d_ws;
    float* h   = acc + ND;
    float* xs1 = h + ND;
    float* xs2 = xs1 + ND;
    float* xs3 = xs2 + ND;
    const float* xs[4] = { x, xs1, xs2, xs3 };

    const dim3 gemmGrid(N / 32), gemmBlock(256);
    const int edgeBlocks = 2048;                 // 16384 warps
    const int n4    = ND / 4;
    const int zgrid = (n4 + 255) / 256;
    const int egrid = (ND + 255) / 256;

    for (int t = 0; t < 4; ++t) {
        zero_f4<<<zgrid, 256, 0, stream>>>((float4*)acc, n4);

        for (int e = 0; e < 3; ++e) {
            const float* W = W_edge + (size_t)(t * 3 + e) * DD * DD;
            const float* b = b_edge + (size_t)(t * 3 + e) * DD;
            gemm_bias_wmma<<<gemmGrid, gemmBlock, 0, stream>>>(xs[t], W, b, h);
            edge_scatter<<<edgeBlocks, 256, 0, stream>>>(h, acc, src, dst, attr, E, /*etype*/1, e);
        }

        for (int k = 2; k <= t + 1; ++k) {
            const int delay = k - 1;             // NU = 1
            const float* W = W_hop + (size_t)(t * 3 + (k - 2)) * DD * DD;
            const float* b = b_hop + (size_t)(t * 3 + (k - 2)) * DD;
            gemm_bias_wmma<<<gemmGrid, gemmBlock, 0, stream>>>(xs[t - delay], W, b, h);
            edge_scatter<<<edgeBlocks, 256, 0, stream>>>(h, acc, src, dst, attr, E, /*hop*/0, k);
        }

        float* xnext = (t < 3) ? const_cast<float*>(xs[t + 1]) : out;
        relu_add<<<egrid, 256, 0, stream>>>(xs[t], acc, xnext, ND);
    }
}